// HDGE_1709396984611
// MI455X (gfx1250) — compile-verified
//
#include <hip/hip_runtime.h>
#include <hip/hip_bf16.h>

typedef __attribute__((ext_vector_type(2))) float v2f;
typedef __attribute__((ext_vector_type(8))) float v8f;

#define N_      256
#define D_      512
#define C_      10
#define K_      65536
#define NPAD    16
#define LCOLS   (K_ + 1)        // logits row length: 1 + K
#define T_INV   (1.0f / 0.07f)
#define LN2     0.6931471805599453f
#define INV_LN2 1.4426950408889634f
#define EPSN    1e-12f

// ---------------------------------------------------------------------------
// Kernel A: f_logit = x @ W  via V_WMMA_F32_16X16X4_F32.
// One block of 512 threads = 16 waves; wave w computes rows [16w, 16w+16).
// W (512x10) is padded to 512x16 in LDS so the WMMA loop is divergence-free.
// ---------------------------------------------------------------------------
__global__ __launch_bounds__(512) void hdge_gemm_wmma(
    const float* __restrict__ x, const float* __restrict__ W,
    float* __restrict__ f_logit) {
  __shared__ float Wl[D_ * NPAD];
  const int tid = threadIdx.x;
  for (int i = tid; i < D_ * NPAD; i += 512) {
    const int row = i >> 4, col = i & 15;
    Wl[i] = (col < C_) ? W[row * C_ + col] : 0.0f;
  }
  __syncthreads();

  const int wave = tid >> 5;            // M tile id (0..15)
  const int lane = tid & 31;
  const int mn   = lane & 15;           // A row / B column within tile
  const int koff = (lane >> 4) << 1;    // 0 for lanes 0-15, 2 for lanes 16-31
  const int arow = wave * 16 + mn;

  v8f acc = {};
  for (int k = 0; k < D_; k += 4) {
    v2f a, b;
    // A 16x4 f32: VGPR0 = K=koff, VGPR1 = K=koff+1
    a.x = x[arow * D_ + k + koff];
    a.y = x[arow * D_ + k + koff + 1];
    // B 4x16 f32: VGPR0 = row K=koff (N=mn), VGPR1 = row K=koff+1
    b.x = Wl[(k + koff) * NPAD + mn];
    b.y = Wl[(k + koff + 1) * NPAD + mn];
    acc = __builtin_amdgcn_wmma_f32_16x16x4_f32(
        /*neg_a=*/false, a, /*neg_b=*/false, b,
        /*c_mod=*/(short)0, acc, /*reuse_a=*/false, /*reuse_b=*/false);
  }

  // D layout: VGPR v -> M = v (lanes 0-15) / v+8 (lanes 16-31); N = lane&15
  const int mbase = wave * 16 + ((lane >> 4) << 3);
  if (mn < C_) {
#pragma unroll
    for (int v = 0; v < 8; ++v)
      f_logit[(mbase + v) * C_ + mn] = acc[v];
  }
}

// ---------------------------------------------------------------------------
// Kernel B: l_pos -> logits[:, 0].  One thread per row n (trivial cost).
// ---------------------------------------------------------------------------
__global__ void hdge_lpos(const float* __restrict__ f_logit,
                          const float* __restrict__ dist,
                          float* __restrict__ logits) {
  const int n = blockIdx.x * blockDim.x + threadIdx.x;
  if (n >= N_) return;
  float f[C_];
  float ss = 0.0f;
#pragma unroll
  for (int c = 0; c < C_; ++c) { f[c] = f_logit[n * C_ + c]; ss += f[c] * f[c]; }
  const float inv = 1.0f / fmaxf(__builtin_sqrtf(ss), EPSN);
  float v[C_];
  float m = -3.402823466e38f;
#pragma unroll
  for (int c = 0; c < C_; ++c) {
    v[c] = dist[n * C_ + c] * (f[c] * inv);
    m = fmaxf(m, v[c]);
  }
  float se = 0.0f;
#pragma unroll
  for (int c = 0; c < C_; ++c)
    se += __builtin_amdgcn_exp2f((v[c] - m) * INV_LN2);
  logits[(size_t)n * LCOLS] = (m + LN2 * __builtin_amdgcn_logf(se)) * T_INV;
}

// ---------------------------------------------------------------------------
// Kernel C: l_neg -> logits[:, 1:].  One thread per queue column k.
// Normalizes the 10-element column once (reused for all 256 rows), keeps
// dist (10 KB) in LDS (wave-uniform broadcast reads), native exp2/log2.
// ---------------------------------------------------------------------------
__global__ __launch_bounds__(256) void hdge_lneg(
    const float* __restrict__ queue, const float* __restrict__ dist,
    float* __restrict__ logits) {
  __shared__ float dist_s[N_ * C_];
  const int tid = threadIdx.x;
  for (int i = tid; i < N_ * C_; i += 256) dist_s[i] = dist[i];
  __syncthreads();

  const int k = blockIdx.x * 256 + tid;
  float b[C_];
  float ss = 0.0f;
#pragma unroll
  for (int c = 0; c < C_; ++c) {
    const float q = queue[(size_t)c * K_ + k];   // coalesced across k
    b[c] = q;
    ss += q * q;
  }
  const float inv = 1.0f / fmaxf(__builtin_sqrtf(ss), EPSN);
#pragma unroll
  for (int c = 0; c < C_; ++c) b[c] *= inv;

  float* __restrict__ out = logits + 1 + k;
  for (int n = 0; n < N_; ++n) {
    float v[C_];
    float m = -3.402823466e38f;
#pragma unroll
    for (int c = 0; c < C_; ++c) {
      v[c] = dist_s[n * C_ + c] * b[c];
      m = fmaxf(m, v[c]);
    }
    float se = 0.0f;
#pragma unroll
    for (int c = 0; c < C_; ++c)
      se += __builtin_amdgcn_exp2f((v[c] - m) * INV_LN2);
    out[(size_t)n * LCOLS] = (m + LN2 * __builtin_amdgcn_logf(se)) * T_INV;
  }
}

// ---------------------------------------------------------------------------
// Kernel D: new_queue = queue with f_logit.T written at columns [start,+256),
// start = clamp(ptr, 0, K-256)  (dynamic_update_slice semantics).
// ---------------------------------------------------------------------------
__global__ void hdge_queue_update(const float* __restrict__ queue,
                                  const float* __restrict__ f_logit,
                                  const int* __restrict__ ptr_p,
                                  float* __restrict__ new_queue) {
  const int idx = blockIdx.x * blockDim.x + threadIdx.x;
  if (idx >= C_ * K_) return;
  const int ptr = *ptr_p;
  const int start = min(max(ptr, 0), K_ - N_);
  const int c = idx >> 16;          // idx = c*K_ + k, K_ = 65536
  const int k = idx & (K_ - 1);
  float val = queue[idx];
  const int off = k - start;
  if (off >= 0 && off < N_) val = f_logit[off * C_ + c];
  new_queue[idx] = val;
}

// ---------------------------------------------------------------------------
extern "C" void kernel_launch(void* const* d_in, const int* in_sizes, int n_in,
                              void* d_out, int out_size, void* d_ws, size_t ws_size,
                              hipStream_t stream) {
  const float* x     = (const float*)d_in[0];   // (256, 512)
  const float* W     = (const float*)d_in[1];   // (512, 10)
  const float* dist  = (const float*)d_in[2];   // (256, 10)
  const float* queue = (const float*)d_in[3];   // (10, 65536)
  const int*   ptr_p = (const int*)d_in[4];     // scalar

  float* out       = (float*)d_out;
  float* logits    = out;                                   // 256 * 65537
  float* f_logit   = out + (size_t)N_ * LCOLS;              // 256 * 10
  float* new_queue = f_logit + (size_t)N_ * C_;             // 10 * 65536

  // A: WMMA GEMM (16 waves, one block)
  hdge_gemm_wmma<<<1, 512, 0, stream>>>(x, W, f_logit);

  // B: l_pos column (depends on f_logit; stream-ordered)
  hdge_lpos<<<1, 256, 0, stream>>>(f_logit, dist, logits);

  // C: l_neg body — 65536 threads, one per queue column
  hdge_lneg<<<K_ / 256, 256, 0, stream>>>(queue, dist, logits);

  // D: circular-buffer enqueue (depends on f_logit; stream-ordered)
  hdge_queue_update<<<(C_ * K_ + 255) / 256, 256, 0, stream>>>(
      queue, f_logit, ptr_p, new_queue);
}